// SimpleSmoothParticleNet_88708254532097
// MI455X (gfx1250) — compile-verified
//
#include <hip/hip_runtime.h>

typedef float v2f __attribute__((ext_vector_type(2)));
typedef float v8f __attribute__((ext_vector_type(8)));

#define B_     2
#define N_     4096
#define C_     16
#define O_     16
#define KS_    27
#define TILES_ (N_ / 16)      // 256 query tiles of 16
#define CHUNK_ 128            // source points staged in LDS per iteration
#define INVR_  10.0f          // 1/RADIUS
#define DIL_   0.05f
#define EPS_   1e-12f

// out[b,i,o] = bias[o]
__global__ void convsp_init_out(const float* __restrict__ bias, float* __restrict__ out) {
  int idx = blockIdx.x * blockDim.x + threadIdx.x;
  if (idx < B_ * N_ * O_) out[idx] = bias[idx & (O_ - 1)];
}

// G[b,k,j,o] = (1/density[b,j]) * sum_c data[b,j,c] * weight[o,c,k]
__global__ void convsp_precompute_g(const float* __restrict__ data,
                                    const float* __restrict__ density,
                                    const float* __restrict__ weight,
                                    float* __restrict__ g) {
  int idx = blockIdx.x * blockDim.x + threadIdx.x;   // (b*N + j)*O + o
  if (idx >= B_ * N_ * O_) return;
  int o  = idx & (O_ - 1);
  int bj = idx >> 4;                                  // b*N + j
  int j  = bj & (N_ - 1);
  int b  = bj >> 12;
  float invd = 1.0f / density[bj];
  float dreg[C_];
#pragma unroll
  for (int c = 0; c < C_; ++c) dreg[c] = data[bj * C_ + c] * invd;
  for (int k = 0; k < KS_; ++k) {
    float acc = 0.0f;
#pragma unroll
    for (int c = 0; c < C_; ++c)
      acc = fmaf(dreg[c], weight[(o * C_ + c) * KS_ + k], acc);
    g[((b * KS_ + k) * N_ + j) * O_ + o] = acc;
  }
}

// SPH kernel weight: w = max(1 - d/R, 0)^3, d = sqrt(max(d2, EPS)).
// Raw v_sqrt_f32 — inputs are O(1) and clamped away from denorms.
__device__ __forceinline__ float sph_w(float d2) {
  float d = __builtin_amdgcn_sqrtf(fmaxf(d2, EPS_));
  float t = fmaxf(fmaf(-INVR_, d, 1.0f), 0.0f);
  return t * t * t;
}

// Main: per (b,k,tile): acc[16x16] = sum_j w_k(i,j) * G[b,k,j,:] via V_WMMA_F32_16X16X4_F32
__global__ void __launch_bounds__(128)
convsp_main(const float* __restrict__ locs, const float* __restrict__ g,
            float* __restrict__ out) {
  __shared__ float s_loc[CHUNK_][4];        // xyz, padded
  __shared__ float s_g[CHUNK_][O_ + 1];     // padded to 17 to dodge bank conflicts

  const int bid        = blockIdx.x;
  const int tile_group = bid % (TILES_ / 4);
  const int k          = (bid / (TILES_ / 4)) % KS_;
  const int b          = bid / ((TILES_ / 4) * KS_);

  const int lane  = threadIdx.x;            // 0..31
  const int wave  = threadIdx.y;            // 0..3  -> tile within group
  const int tid   = wave * 32 + lane;       // 0..127 for cooperative staging
  const int tile  = tile_group * 4 + wave;
  const int row   = lane & 15;              // A: M-row, B: N-col (o)
  const int jslot = (lane >> 4) << 1;       // K-slots {0,1} or {2,3}

  // kernel-cell offset (itertools.product order: x slowest)
  const float offx = (float)(k / 9 - 1) * DIL_;
  const float offy = (float)((k / 3) % 3 - 1) * DIL_;
  const float offz = (float)(k % 3 - 1) * DIL_;

  // query center for this lane's M-row (lanes l and l+16 duplicate the row)
  const int i = tile * 16 + row;
  const float* lp = locs + ((size_t)b * N_ + i) * 3;
  const float qx = lp[0] + offx, qy = lp[1] + offy, qz = lp[2] + offz;

  const float* gp = g + ((size_t)(b * KS_ + k) * N_) * O_;
  const float* lb = locs + (size_t)b * N_ * 3;

  v8f acc = {};
  for (int j0 = 0; j0 < N_; j0 += CHUNK_) {
    // cooperative staging across all 4 waves (all 128 threads active -> EXEC full)
    for (int t = tid; t < CHUNK_; t += 128) {
      s_loc[t][0] = lb[(j0 + t) * 3 + 0];
      s_loc[t][1] = lb[(j0 + t) * 3 + 1];
      s_loc[t][2] = lb[(j0 + t) * 3 + 2];
    }
    for (int t = tid; t < CHUNK_ * O_; t += 128) {
      s_g[t >> 4][t & 15] = gp[(size_t)(j0 + (t >> 4)) * O_ + (t & 15)];
    }
    __syncthreads();

    // software pipeline: prefetch iteration jj's operands one step ahead so the
    // ds_load -> s_wait_dscnt latency is covered by the previous iteration's VALU.
    float p0x = s_loc[jslot][0],     p0y = s_loc[jslot][1],     p0z = s_loc[jslot][2];
    float p1x = s_loc[jslot + 1][0], p1y = s_loc[jslot + 1][1], p1z = s_loc[jslot + 1][2];
    v2f bm;  bm.x = s_g[jslot][row]; bm.y = s_g[jslot + 1][row];

#pragma unroll 8
    for (int jj = 0; jj < CHUNK_; jj += 4) {
      // capture current operands
      const float cx0 = p0x, cy0 = p0y, cz0 = p0z;
      const float cx1 = p1x, cy1 = p1y, cz1 = p1z;
      v2f bcur = bm;

      // prefetch next quad (wraps on last iteration; values discarded)
      const int ja = ((jj + 4) & (CHUNK_ - 1)) + jslot;
      p0x = s_loc[ja][0];     p0y = s_loc[ja][1];     p0z = s_loc[ja][2];
      p1x = s_loc[ja + 1][0]; p1y = s_loc[ja + 1][1]; p1z = s_loc[ja + 1][2];
      bm.x = s_g[ja][row];    bm.y = s_g[ja + 1][row];

      // pair 0: (row, jj+jslot)
      float dx = qx - cx0, dy = qy - cy0, dz = qz - cz0;
      float t0 = sph_w(fmaf(dx, dx, fmaf(dy, dy, dz * dz)));
      // pair 1: (row, jj+jslot+1)
      dx = qx - cx1; dy = qy - cy1; dz = qz - cz1;
      float t1 = sph_w(fmaf(dx, dx, fmaf(dy, dy, dz * dz)));

      v2f a; a.x = t0; a.y = t1;                     // A 16x4 f32 layout
      acc = __builtin_amdgcn_wmma_f32_16x16x4_f32(
          false, a, false, bcur, (short)0, acc, false, false);
    }
    __syncthreads();
  }

  // D layout: VGPR v -> M = v + 8*(lane/16), N = lane%16; accumulate over k via atomics
  const int ibase = tile * 16 + ((lane >> 4) << 3);
#pragma unroll
  for (int v = 0; v < 8; ++v) {
    atomicAdd(&out[((size_t)b * N_ + ibase + v) * O_ + row], acc[v]);
  }
}

extern "C" void kernel_launch(void* const* d_in, const int* in_sizes, int n_in,
                              void* d_out, int out_size, void* d_ws, size_t ws_size,
                              hipStream_t stream) {
  const float* locs    = (const float*)d_in[0];
  const float* data    = (const float*)d_in[1];
  const float* density = (const float*)d_in[2];
  const float* weight  = (const float*)d_in[3];
  const float* bias    = (const float*)d_in[4];
  float* out = (float*)d_out;
  float* g   = (float*)d_ws;   // needs B*K*N*O*4 = ~14.2 MB of workspace

  convsp_init_out<<<(B_ * N_ * O_ + 255) / 256, 256, 0, stream>>>(bias, out);
  convsp_precompute_g<<<(B_ * N_ * O_ + 255) / 256, 256, 0, stream>>>(data, density, weight, g);

  dim3 blk(32, 4);   // 4 waves share one (b,k) LDS stage, each owns one 16-query tile
  convsp_main<<<B_ * KS_ * (TILES_ / 4), blk, 0, stream>>>(locs, g, out);
}